// LilletLayer_17600775979100
// MI455X (gfx1250) — compile-verified
//
#include <hip/hip_runtime.h>
#include <math.h>

// Problem constants (from reference)
#define B_    128
#define N_    512
#define H_    8
#define K_    6
#define R_    50
#define HID_  128
#define A_    36              // K*K
#define CHUNK_ 1800           // A_*R_  (features per (h,a) chunk)
#define NHA_  288             // H_*A_
#define NSUB_ 3               // split each chunk 3-ways (c-ranges; 600 % 4 == 0)
#define SUBF_ 600             // CHUNK_/NSUB_
#define NPART_ (NHA_*NSUB_)   // 864 partial blocks

typedef float v2f __attribute__((ext_vector_type(2)));
typedef float v8f __attribute__((ext_vector_type(8)));

// Workspace layout (float elements)
#define XC_OFF   0
#define XC_SZ    (B_*H_*K_*3)                 // 18,432
#define BAS_OFF  (XC_OFF + XC_SZ)
#define BAS_SZ   (H_*A_*R_*3*B_)              // 5,529,600  (layout [h][a][n][d][b])
#define PART_OFF (BAS_OFF + BAS_SZ)
#define PART_SZ  ((size_t)NPART_*B_*HID_)     // 14,155,776

// ---------------- Stage 1: xc[b][h][k][d] = sum_n W_map[h][k][n] * x[b][n][d]
__global__ void k_xc(const float* __restrict__ x, const float* __restrict__ Wm,
                     float* __restrict__ ws) {
  int b = blockIdx.x;
  int t = threadIdx.x;
  if (t >= H_ * K_ * 3) return;
  int d = t % 3, hk = t / 3;                  // hk = h*6+k
  const float* wrow = Wm + hk * N_;
  const float* xb = x + (size_t)b * N_ * 3 + d;
  // 4 independent accumulators to break the dependent-FMA latency chain
  float s0 = 0.f, s1 = 0.f, s2 = 0.f, s3 = 0.f;
  for (int n = 0; n < N_; n += 4) {
    s0 += wrow[n]     * xb[n * 3];
    s1 += wrow[n + 1] * xb[(n + 1) * 3];
    s2 += wrow[n + 2] * xb[(n + 2) * 3];
    s3 += wrow[n + 3] * xb[(n + 3) * 3];
  }
  ws[XC_OFF + (b * (H_ * K_) + hk) * 3 + d] = (s0 + s1) + (s2 + s3);
}

// ---------------- Stage 2: basis[h][a][r][d][b]  (b innermost for coalescing)
__global__ void k_basis(const float* __restrict__ means, const float* __restrict__ betas,
                        float* __restrict__ ws) {
  int tid = blockIdx.x * blockDim.x + threadIdx.x;
  if (tid >= H_ * A_ * R_ * B_) return;
  int b  = tid & (B_ - 1);
  int q  = tid >> 7;                          // q = (h*36+a)*50 + r
  int r  = q % R_;
  int ar = q / R_;                            // h*36 + a
  int a  = ar % A_;
  int h  = ar / A_;
  int i  = a / K_;
  int j  = a % K_;
  const float* xci = ws + XC_OFF + ((b * H_ + h) * K_ + i) * 3;
  const float* xcj = ws + XC_OFF + ((b * H_ + h) * K_ + j) * 3;
  float dx = xci[0] - xcj[0];
  float dy = xci[1] - xcj[1];
  float dz = xci[2] - xcj[2];
  float d2 = dx * dx + dy * dy + dz * dz;
  float dn = d2 > 0.f ? sqrtf(d2) : 0.f;
  float cut = dn < 5.0f ? 0.5f * (__cosf(dn * 0.62831853071795864769f) + 1.f) : 0.f;
  float t  = __expf(-dn) - means[r];
  float sm = cut * __expf(-betas[r] * t * t);
  float inv = dn + 1e-6f;
  sm = sm / (inv * inv);
  float* out = ws + BAS_OFF + (size_t)q * 3 * B_ + b;    // [q][d][b]
  out[0]        = dx * sm;
  out[B_]       = dy * sm;
  out[2 * B_]   = dz * sm;
}

// ---------------- Stage 3: fused att + GEMM partials via V_WMMA_F32_16X16X4_F32
// Block (ha, ksub): features f in [ksub*600, ksub*600+600) of chunk ha = h*36+a.
// 8 waves; wave w owns batch rows b = w*16..w*16+15 and ALL 128 output cols
// (8 WMMA accumulator tiles) so W1 is streamed exactly once from HBM.
// W1 is a pure stream -> non-temporal loads + explicit prefetch, keeping the
// 192MB L2 for the reused basis (22MB) and partials.
__global__ void __launch_bounds__(256)
k_gemm(const float* __restrict__ W1, const float* __restrict__ bas,
       float* __restrict__ parts) {
  const int ha    = blockIdx.x;               // 0..287
  const int ksub  = blockIdx.y;               // 0..2
  const int ch    = ha * NSUB_ + ksub;
  const int wave  = threadIdx.x >> 5;
  const int lane  = threadIdx.x & 31;
  const int mrow  = lane & 15;                // A row within tile / B col within tile
  const int khalf = lane >> 4;                // K pair select (ISA A/B 16x4 layout)
  const int b     = (wave << 4) + mrow;       // batch row fed by this lane
  const int h     = ha / A_;
  const int a     = ha % A_;

  // basis slice for this h, offset to this lane's b
  const float* basH = bas + (size_t)h * (A_ * R_ * 3 * B_) + b;
  const int an = a * (R_ * 3 * B_);           // a-side row offset
  // W1 rows for this feature chunk, at this lane's base column
  const float* W1c = W1 + (size_t)ha * CHUNK_ * HID_ + mrow;

  v8f acc[8];
#pragma unroll
  for (int t = 0; t < 8; ++t) acc[t] = (v8f){0.f,0.f,0.f,0.f,0.f,0.f,0.f,0.f};

  const int fbeg = ksub * SUBF_;
  const int fend = fbeg + SUBF_;
  for (int ks = fbeg; ks < fend; ks += 4) {
    const int f0 = ks + (khalf << 1);
    // ---- A fragment: two on-the-fly att elements (3-dot of basis vectors)
    v2f afrag;
#pragma unroll
    for (int e = 0; e < 2; ++e) {
      const int f = f0 + e;
      const int c = f / R_;
      const int n = f - c * R_;
      const int no = n * (3 * B_);
      const int co = c * (R_ * 3 * B_) + no;
      float s = basH[an + no]            * basH[co];
      s      += basH[an + no + B_]       * basH[co + B_];
      s      += basH[an + no + 2 * B_]   * basH[co + 2 * B_];
      afrag[e] = s;
    }
    // ---- B fragments: W1 rows f0, f0+1 for all 8 column tiles (coalesced 64B)
    const float* W1r0 = W1c + (size_t)f0 * HID_;
    const float* W1r1 = W1r0 + HID_;
    // stream-prefetch 16 rows (8KB) ahead; uniform guard keeps it in-bounds
    if (ks + 20 <= fend) {
      __builtin_prefetch((const void*)(W1r0 + 16 * HID_), 0, 0);
    }
#pragma unroll
    for (int t = 0; t < 8; ++t) {
      v2f bfrag;
      bfrag[0] = __builtin_nontemporal_load(&W1r0[t * 16]);
      bfrag[1] = __builtin_nontemporal_load(&W1r1[t * 16]);
      acc[t] = __builtin_amdgcn_wmma_f32_16x16x4_f32(
          false, afrag, false, bfrag, (short)0, acc[t], false, false);
    }
  }

  // ---- store partials: parts[ch][row][col], row = wave*16 + khalf*8 + v
  float* pb = parts + ((size_t)ch * B_ + (wave << 4) + (khalf << 3)) * HID_ + mrow;
#pragma unroll
  for (int t = 0; t < 8; ++t) {
#pragma unroll
    for (int v = 0; v < 8; ++v) {
      pb[(size_t)v * HID_ + t * 16] = acc[t][v];
    }
  }
}

// ---------------- Stage 4: deterministic split-K reduce + bias + SiLU + W2 dot
__global__ void k_final(const float* __restrict__ b1, const float* __restrict__ W2,
                        const float* __restrict__ b2, const float* __restrict__ parts,
                        float* __restrict__ out) {
  __shared__ float red[HID_];
  const int b = blockIdx.x;
  const int j = threadIdx.x;
  float s = b1[j];
  const float* p = parts + (size_t)b * HID_ + j;
  for (int ch = 0; ch < NPART_; ++ch) s += p[(size_t)ch * B_ * HID_];
  float act = s / (1.f + __expf(-s));         // silu
  red[j] = act * W2[j];
  __syncthreads();
  if (j == 0) {
    float o = b2[0];
    for (int k = 0; k < HID_; ++k) o += red[k];
    out[b] = o;
  }
}

extern "C" void kernel_launch(void* const* d_in, const int* in_sizes, int n_in,
                              void* d_out, int out_size, void* d_ws, size_t ws_size,
                              hipStream_t stream) {
  (void)in_sizes; (void)n_in; (void)out_size; (void)ws_size;
  const float* x     = (const float*)d_in[0];
  const float* Wmap  = (const float*)d_in[1];
  const float* means = (const float*)d_in[2];
  const float* betas = (const float*)d_in[3];
  const float* W1    = (const float*)d_in[4];
  const float* b1    = (const float*)d_in[5];
  const float* W2    = (const float*)d_in[6];
  const float* b2    = (const float*)d_in[7];
  float* ws  = (float*)d_ws;
  float* out = (float*)d_out;

  k_xc<<<B_, 160, 0, stream>>>(x, Wmap, ws);

  const int bt = H_ * A_ * R_ * B_;            // 1,843,200 threads
  k_basis<<<(bt + 255) / 256, 256, 0, stream>>>(means, betas, ws);

  dim3 g(NHA_, NSUB_);                          // 288 x 3 blocks
  k_gemm<<<g, 256, 0, stream>>>(W1, ws + BAS_OFF, ws + PART_OFF);

  k_final<<<B_, HID_, 0, stream>>>(b1, W2, b2, ws + PART_OFF, out);
}